// LlamaAttention_58480274703236
// MI455X (gfx1250) — compile-verified
//
#include <hip/hip_runtime.h>
#include <hip/hip_bf16.h>
#include <math.h>

// ---- problem constants (from reference) ----
#define HB    32        // batch
#define HH    32        // q heads
#define HKV   8         // kv heads
#define HG    4         // q heads per kv head
#define HD    128       // head dim
#define HS    2048      // seq len
#define HHID  4096      // hidden = HH*HD
#define HQKV  6144      // (HH + 2*HKV)*HD
#define RSCALE 0.08838834764831845f  // D^-0.5
#define RTHETA 10000.0f
#define NEG_BIG (-3.402823466e38f)

typedef __attribute__((ext_vector_type(2))) float v2f;
typedef __attribute__((ext_vector_type(8))) float v8f;

// ---------------------------------------------------------------------------
// fp32 WMMA GEMM:  C[32 x N] = A[32 x K] * Bw^T, Bw is (N x K) row-major.
// One 16x16 C tile per wave via V_WMMA_F32_16X16X4_F32.
//   tiles = 2 (m) * N/16 (n); grid.x * 8 waves == tiles.
// ---------------------------------------------------------------------------
__global__ __launch_bounds__(256) void gemm32_wmma_f32(
    const float* __restrict__ A, const float* __restrict__ Bw,
    float* __restrict__ C, int K, int N)
{
    const int lane = threadIdx.x & 31;
    const int wave = threadIdx.x >> 5;
    const int tile = blockIdx.x * 8 + wave;
    const int mtile = tile & 1;        // 0..1
    const int ntile = tile >> 1;       // 0..N/16-1

    const int r15   = lane & 15;
    const int khalf = (lane >> 4) << 1;   // 0 or 2

    // A fragment: lane holds (M = mtile*16 + r15, K = k0 + khalf + {0,1})
    const float* aptr = A  + (mtile * 16 + r15) * (size_t)K + khalf;
    // B fragment: lane holds (N = ntile*16 + r15, K = k0 + khalf + {0,1})
    const float* bptr = Bw + (ntile * 16 + r15) * (size_t)K + khalf;

    v8f acc = {0.f, 0.f, 0.f, 0.f, 0.f, 0.f, 0.f, 0.f};

    const int CH = 64;  // 16 WMMAs per chunk
    for (int k0 = 0; k0 < K; k0 += CH) {
        // stream next chunk into cache hierarchy (global_prefetch_b8)
        __builtin_prefetch(bptr + k0 + CH, 0, 1);
        __builtin_prefetch(aptr + k0 + CH, 0, 1);
#pragma unroll
        for (int kk = 0; kk < CH; kk += 4) {
            v2f a = *(const v2f*)(aptr + k0 + kk);
            v2f b = *(const v2f*)(bptr + k0 + kk);
            acc = __builtin_amdgcn_wmma_f32_16x16x4_f32(
                false, a, false, b, (short)0, acc, false, false);
        }
    }

    // C/D layout: VGPR r -> (M = r + 8*(lane>=16), N = lane&15)
    const int mbase = mtile * 16 + ((lane >> 4) << 3);
    const int n     = ntile * 16 + r15;
#pragma unroll
    for (int r = 0; r < 8; ++r)
        C[(size_t)(mbase + r) * N + n] = acc[r];
}

// ---------------------------------------------------------------------------
// RoPE in-place on qkv scratch. Heads 0..31 = q, 32..39 = k (offset h*128
// works for both since k starts at 4096). 64 threads: one rotation pair each.
// ---------------------------------------------------------------------------
__global__ __launch_bounds__(64) void rope_kernel(
    float* __restrict__ qkv, const int* __restrict__ positions)
{
    const int b = blockIdx.x / (HH + HKV);
    const int h = blockIdx.x % (HH + HKV);
    float* row = qkv + (size_t)b * HQKV + h * HD;

    const int i = threadIdx.x;           // 0..63
    const float pos = (float)positions[b];
    const float inv_freq = __expf(-((float)i / 64.0f) * __logf(RTHETA));
    const float f = pos * inv_freq;
    const float c = __cosf(f), s = __sinf(f);

    const float x1 = row[i];
    const float x2 = row[i + 64];
    row[i]      = x1 * c - x2 * s;
    row[i + 64] = x2 * c + x1 * s;
}

// ---------------------------------------------------------------------------
// block reduce (8 waves of 32)
// ---------------------------------------------------------------------------
__device__ __forceinline__ float blk_reduce(float v, float* s_red, bool is_max)
{
#pragma unroll
    for (int off = 16; off; off >>= 1) {
        float o = __shfl_xor(v, off, 32);
        v = is_max ? fmaxf(v, o) : (v + o);
    }
    const int wv = threadIdx.x >> 5, ln = threadIdx.x & 31;
    if (ln == 0) s_red[wv] = v;
    __syncthreads();
    if (threadIdx.x < 32) {
        float x = (ln < 8) ? s_red[ln] : (is_max ? NEG_BIG : 0.0f);
#pragma unroll
        for (int off = 4; off; off >>= 1) {
            float o = __shfl_xor(x, off, 32);
            x = is_max ? fmaxf(x, o) : (x + o);
        }
        if (ln == 0) s_red[0] = x;
    }
    __syncthreads();
    float r = s_red[0];
    __syncthreads();
    return r;
}

// ---------------------------------------------------------------------------
// GQA decode attention. One block per (b, kv-head). 256 threads.
// ---------------------------------------------------------------------------
__global__ __launch_bounds__(256) void attn_kernel(
    const float* __restrict__ qkv, const float* __restrict__ k_cache,
    const float* __restrict__ v_cache, const int* __restrict__ positions,
    float* __restrict__ attn_out)
{
    __shared__ float s_q[HG * HD];          // scaled q, 2KB
    __shared__ float s_sc[HG * HS];         // scores/probs, 32KB
    __shared__ float s_red[8];
    __shared__ float s_stat[2 * HG];        // [0..3]=max, [4..7]=1/sum

    const int b   = blockIdx.x >> 3;
    const int kvh = blockIdx.x & 7;
    const int t   = threadIdx.x;
    int pos = positions[b];
    if (pos < 0) pos = 0;
    if (pos >= HS) pos = HS - 1;

    const float* k_new = qkv + (size_t)b * HQKV + HHID + kvh * HD;
    const float* v_new = qkv + (size_t)b * HQKV + HHID + HKV * HD + kvh * HD;

    // load q (scaled)
#pragma unroll
    for (int i = t; i < HG * HD; i += 256)
        s_q[i] = qkv[(size_t)b * HQKV + (kvh * HG) * HD + i] * RSCALE;
    // init scores to -inf (masked region stays -inf)
    for (int i = t; i < HG * HS; i += 256)
        s_sc[i] = NEG_BIG;
    __syncthreads();

    // ---- scores: 4 lanes per key row, each lane covers 32 dims ----
    const int sub = t & 3;          // dim quarter
    const int sid = t >> 2;         // 0..63
    for (int s = sid; s <= pos; s += 64) {
        const float* kp = (s == pos)
            ? k_new
            : k_cache + (((size_t)b * HS + s) * HKV + kvh) * HD;
        float acc[HG] = {0.f, 0.f, 0.f, 0.f};
        const int d0 = sub * 32;
#pragma unroll
        for (int d = 0; d < 32; d += 4) {
            const float4 kv = *(const float4*)(kp + d0 + d);
#pragma unroll
            for (int g = 0; g < HG; ++g) {
                const float4 qv = *(const float4*)(&s_q[g * HD + d0 + d]);
                acc[g] += qv.x * kv.x + qv.y * kv.y + qv.z * kv.z + qv.w * kv.w;
            }
        }
#pragma unroll
        for (int g = 0; g < HG; ++g) {
            acc[g] += __shfl_xor(acc[g], 1, 32);
            acc[g] += __shfl_xor(acc[g], 2, 32);
        }
        if (sub == 0) {
#pragma unroll
            for (int g = 0; g < HG; ++g) s_sc[g * HS + s] = acc[g];
        }
    }
    __syncthreads();

    // ---- softmax per g ----
    for (int g = 0; g < HG; ++g) {
        float m = NEG_BIG;
        for (int s = t; s < HS; s += 256) m = fmaxf(m, s_sc[g * HS + s]);
        m = blk_reduce(m, s_red, true);
        if (t == 0) s_stat[g] = m;
        float part = 0.f;
        for (int s = t; s < HS; s += 256) {
            float p = __expf(s_sc[g * HS + s] - m);
            s_sc[g * HS + s] = p;
            part += p;
        }
        __syncthreads();
        float sum = blk_reduce(part, s_red, false);
        if (t == 0) s_stat[HG + g] = 1.0f / sum;
        __syncthreads();
    }

    // ---- probs @ V : thread owns (g, g+2) x one dim ----
    const int g0 = t >> 7;          // 0 or 1
    const int d  = t & 127;
    float accA = 0.f, accB = 0.f;
    for (int s = 0; s < pos; ++s) {
        const float vv = v_cache[(((size_t)b * HS + s) * HKV + kvh) * HD + d];
        accA += s_sc[g0 * HS + s]       * vv;
        accB += s_sc[(g0 + 2) * HS + s] * vv;
    }
    { // current position uses freshly-projected v
        const float vv = v_new[d];
        accA += s_sc[g0 * HS + pos]       * vv;
        accB += s_sc[(g0 + 2) * HS + pos] * vv;
    }
    float* outp = attn_out + (size_t)b * HHID + kvh * HG * HD;
    outp[ g0      * HD + d] = accA * s_stat[HG + g0];
    outp[(g0 + 2) * HD + d] = accB * s_stat[HG + g0 + 2];
}

// ---------------------------------------------------------------------------
extern "C" void kernel_launch(void* const* d_in, const int* in_sizes, int n_in,
                              void* d_out, int out_size, void* d_ws, size_t ws_size,
                              hipStream_t stream)
{
    const float* hidden  = (const float*)d_in[0];   // (32, 4096)
    const float* w_qkv   = (const float*)d_in[1];   // (6144, 4096)
    const float* w_o     = (const float*)d_in[2];   // (4096, 4096)
    const float* k_cache = (const float*)d_in[3];   // (32, 2048, 8, 128)
    const float* v_cache = (const float*)d_in[4];   // (32, 2048, 8, 128)
    const int*   pos     = (const int*)d_in[5];     // (32,)
    float*       out     = (float*)d_out;           // (32, 4096)

    float* qkv_ws  = (float*)d_ws;                         // 32*6144 floats
    float* attn_ws = qkv_ws + (size_t)HB * HQKV;           // 32*4096 floats

    // 1) qkv = hidden @ w_qkv.T   : tiles = 2 * 6144/16 = 768 -> 96 blocks
    gemm32_wmma_f32<<<96, 256, 0, stream>>>(hidden, w_qkv, qkv_ws, HHID, HQKV);

    // 2) RoPE on q and k
    rope_kernel<<<HB * (HH + HKV), 64, 0, stream>>>(qkv_ws, pos);

    // 3) attention -> attn_ws
    attn_kernel<<<HB * HKV, 256, 0, stream>>>(qkv_ws, k_cache, v_cache, pos,
                                              attn_ws);

    // 4) out = attn @ w_o.T       : tiles = 2 * 4096/16 = 512 -> 64 blocks
    gemm32_wmma_f32<<<64, 256, 0, stream>>>(attn_ws, w_o, out, HHID, HHID);
}